// NCC_vfa_730144441169
// MI455X (gfx1250) — compile-verified
//
#include <hip/hip_runtime.h>
#include <stdint.h>

// ---------------------------------------------------------------------------
// Multi-scale 3D NCC loss, fully fused per scale.
//
// MI455X reasoning: the whole problem (~90 MB across the 3-level pyramid)
// fits in the 192 MB L2; arithmetic is ~3 GFLOP total. This is a pure
// data-movement problem, so the kernel fuses all three separable box-filter
// axes + the cc formula + the reduction into ONE LDS-tiled kernel per scale,
// staged with CDNA5 async global->LDS copies (ASYNCcnt) so no 5-field
// intermediate volumes ever hit HBM. WMMA is deliberately not used: there is
// no matmul here, and f16/bf16 WMMA would destroy precision in the
// E[x^2]-mu^2 cancellation.
// ---------------------------------------------------------------------------

#define NCC_NTHREADS 256

namespace {

constexpr int TW = 8, TH = 8, TD = 8;   // output tile per block

__device__ __forceinline__ unsigned lds_lo32(const void* p) {
  // low 32 bits of a generic (flat) shared-memory address == wave-relative
  // LDS byte offset on gfx1250 (aperture bits live in the high dword).
  return (unsigned)(uintptr_t)p;
}

__device__ __forceinline__ void async_gl_to_lds_b32(unsigned lds_addr, const float* g) {
  // CDNA5 async copy: copies 4 bytes from the global address in vaddr to the LDS address in vdst, tracked with ASYNCcnt.
  asm volatile("global_load_async_to_lds_b32 %0, %1, off"
               :: "v"(lds_addr), "v"(g)
               : "memory");
}

__device__ __forceinline__ void wait_asynccnt0() {
  asm volatile("s_wait_asynccnt 0" ::: "memory");
}

// One fused NCC scale: box-filter (win = 2P+1, zero 'same' padding) of
// {I, J, I*I, J*J, I*J} along W,H,D inside LDS, then cc per voxel, then a
// deterministic per-block partial sum.
template <int P>
__global__ __launch_bounds__(NCC_NTHREADS)
void ncc_scale_kernel(const float* __restrict__ I, const float* __restrict__ J,
                      int D, int H, int W, int dblocks,
                      float* __restrict__ partials) {
  constexpr int WIN = 2 * P + 1;
  constexpr int HD  = TD + 2 * P;
  constexpr int HH  = TH + 2 * P;
  constexpr int HW  = TW + 2 * P;
  constexpr int SW  = HW + 1;   // padded w-stride (17 coprime 64 -> no bank conflicts)
  constexpr int TWP = TW + 1;   // padded w-stride for temporaries
  constexpr float INVW = 1.0f / (float)WIN;

  __shared__ float sI[HD * HH * SW];
  __shared__ float sJ[HD * HH * SW];
  __shared__ float tW[HD * HH * TWP];
  __shared__ float tH[HD * TH * TWP];
  __shared__ float red[NCC_NTHREADS];

  const int tid = threadIdx.x;
  const int wb  = blockIdx.x * TW;
  const int hb  = blockIdx.y * TH;
  const int n   = blockIdx.z / dblocks;
  const int db  = (blockIdx.z % dblocks) * TD;

  const size_t plane = (size_t)H * (size_t)W;
  const float* In = I + (size_t)n * (size_t)D * plane;
  const float* Jn = J + (size_t)n * (size_t)D * plane;

  // ---- stage halo tile (zero padded outside the volume) via async copies ----
  for (int idx = tid; idx < HD * HH * HW; idx += NCC_NTHREADS) {
    const int lw = idx % HW;
    const int r  = idx / HW;
    const int lh = r % HH;
    const int ld = r / HH;
    const int gw = wb + lw - P;
    const int gh = hb + lh - P;
    const int gd = db + ld - P;
    const int loff = (ld * HH + lh) * SW + lw;
    if ((unsigned)gw < (unsigned)W && (unsigned)gh < (unsigned)H &&
        (unsigned)gd < (unsigned)D) {
      const size_t g = ((size_t)gd * (size_t)H + (size_t)gh) * (size_t)W + (size_t)gw;
      async_gl_to_lds_b32(lds_lo32(&sI[loff]), In + g);
      async_gl_to_lds_b32(lds_lo32(&sJ[loff]), Jn + g);
    } else {
      // disjoint addresses from the async writes -> no ordering hazard
      sI[loff] = 0.0f;
      sJ[loff] = 0.0f;
    }
  }
  wait_asynccnt0();
  __syncthreads();

  // this thread's two output voxels: (odq, odq+4, oh, ow)
  const int ow  = tid & (TW - 1);
  const int oh  = (tid >> 3) & (TH - 1);
  const int odq = tid >> 6;   // 0..3

  float res[2][5];

#pragma unroll
  for (int f = 0; f < 5; ++f) {
    // ---- W pass: one row per thread, register-staged ----
    for (int r = tid; r < HD * HH; r += NCC_NTHREADS) {
      float rI[HW], rJ[HW];
      const float* pI = &sI[r * SW];
      const float* pJ = &sJ[r * SW];
#pragma unroll
      for (int k = 0; k < HW; ++k) { rI[k] = pI[k]; rJ[k] = pJ[k]; }
#pragma unroll
      for (int wo = 0; wo < TW; ++wo) {
        float s = 0.0f;
#pragma unroll
        for (int k = 0; k < WIN; ++k) {
          const float a = rI[wo + k];
          const float b = rJ[wo + k];
          s += (f == 0) ? a
             : (f == 1) ? b
             : (f == 2) ? a * a
             : (f == 3) ? b * b
                        : a * b;
        }
        tW[r * TWP + wo] = s * INVW;
      }
    }
    __syncthreads();

    // ---- H pass: one (d, w) column per thread, register-staged ----
    for (int c = tid; c < HD * TW; c += NCC_NTHREADS) {
      const int w0 = c & (TW - 1);
      const int d  = c >> 3;
      float colv[HH];
#pragma unroll
      for (int k = 0; k < HH; ++k) colv[k] = tW[(d * HH + k) * TWP + w0];
#pragma unroll
      for (int ho = 0; ho < TH; ++ho) {
        float s = 0.0f;
#pragma unroll
        for (int k = 0; k < WIN; ++k) s += colv[ho + k];
        tH[(d * TH + ho) * TWP + w0] = s * INVW;
      }
    }
    __syncthreads();

    // ---- D pass: straight into registers ----
#pragma unroll
    for (int t = 0; t < 2; ++t) {
      const int od = odq + t * 4;
      float s = 0.0f;
#pragma unroll
      for (int k = 0; k < WIN; ++k)
        s += tH[((od + k) * TH + oh) * TWP + ow];
      res[t][f] = s * INVW;
    }
    __syncthreads();   // protect tW/tH before next field overwrites them
  }

  // ---- cc per voxel + deterministic block reduction ----
  float acc = 0.0f;
#pragma unroll
  for (int t = 0; t < 2; ++t) {
    const int od = odq + t * 4;
    const int gw = wb + ow, gh = hb + oh, gd = db + od;
    if (gw < W && gh < H && gd < D) {
      const float mu2 = res[t][0];            // box(I)
      const float mu1 = res[t][1];            // box(J)
      const float s2  = res[t][2] - mu2 * mu2;
      const float s1  = res[t][3] - mu1 * mu1;
      const float s12 = res[t][4] - mu1 * mu2;
      const float den = fmaxf(s1 * s2, 1.1920928955078125e-07f); // f32 eps
      acc += (s12 * s12) / den;
    }
  }
  red[tid] = acc;
  __syncthreads();
#pragma unroll
  for (int s = NCC_NTHREADS / 2; s > 0; s >>= 1) {
    if (tid < s) red[tid] += red[tid + s];
    __syncthreads();
  }
  if (tid == 0) {
    const int bid = blockIdx.x + gridDim.x * (blockIdx.y + gridDim.y * blockIdx.z);
    partials[bid] = red[0];
  }
}

// avg_pool3d, k=3, stride=2, pad=1, count_include_pad=False
__global__ __launch_bounds__(NCC_NTHREADS)
void avgpool_kernel(const float* __restrict__ x, float* __restrict__ y,
                    int D, int H, int W, int Do, int Ho, int Wo, int N) {
  const size_t total = (size_t)N * Do * Ho * Wo;
  for (size_t i = (size_t)blockIdx.x * blockDim.x + threadIdx.x; i < total;
       i += (size_t)gridDim.x * blockDim.x) {
    const int wo = (int)(i % Wo);
    size_t r = i / Wo;
    const int ho = (int)(r % Ho); r /= Ho;
    const int dd = (int)(r % Do);
    const int n  = (int)(r / Do);
    const int ws = wo * 2 - 1, hs = ho * 2 - 1, zs = dd * 2 - 1;
    const float* xn = x + (size_t)n * D * H * W;
    float s = 0.0f;
    int cnt = 0;
#pragma unroll
    for (int a = 0; a < 3; ++a) {
      const int d = zs + a;
      if ((unsigned)d >= (unsigned)D) continue;
#pragma unroll
      for (int b = 0; b < 3; ++b) {
        const int h = hs + b;
        if ((unsigned)h >= (unsigned)H) continue;
#pragma unroll
        for (int c = 0; c < 3; ++c) {
          const int w = ws + c;
          if ((unsigned)w >= (unsigned)W) continue;
          s += xn[((size_t)d * H + h) * W + w];
          ++cnt;
        }
      }
    }
    y[i] = s / (float)cnt;
  }
}

// Deterministic final reduction: out = sum_s c_s * sum(partials_s)
__global__ __launch_bounds__(NCC_NTHREADS)
void final_reduce_kernel(const float* __restrict__ p0, int n0,
                         const float* __restrict__ p1, int n1,
                         const float* __restrict__ p2, int n2,
                         float c0, float c1, float c2,
                         float* __restrict__ out) {
  __shared__ double red[NCC_NTHREADS];
  double s0 = 0.0, s1 = 0.0, s2 = 0.0;
  for (int i = threadIdx.x; i < n0; i += NCC_NTHREADS) s0 += (double)p0[i];
  for (int i = threadIdx.x; i < n1; i += NCC_NTHREADS) s1 += (double)p1[i];
  for (int i = threadIdx.x; i < n2; i += NCC_NTHREADS) s2 += (double)p2[i];
  red[threadIdx.x] = s0 * (double)c0 + s1 * (double)c1 + s2 * (double)c2;
  __syncthreads();
  for (int s = NCC_NTHREADS / 2; s > 0; s >>= 1) {
    if (threadIdx.x < s) red[threadIdx.x] += red[threadIdx.x + s];
    __syncthreads();
  }
  if (threadIdx.x == 0) out[0] = (float)red[0];
}

inline unsigned cdiv(unsigned a, unsigned b) { return (a + b - 1) / b; }

}  // namespace

extern "C" void kernel_launch(void* const* d_in, const int* in_sizes, int n_in,
                              void* d_out, int out_size, void* d_ws, size_t ws_size,
                              hipStream_t stream) {
  (void)in_sizes; (void)n_in; (void)out_size; (void)ws_size;

  const float* I0 = (const float*)d_in[0];
  const float* J0 = (const float*)d_in[1];
  float* out = (float*)d_out;

  const int N = 2;
  const int D0 = 160, H0 = 192, W0 = 160;   // win 9
  const int D1 = 80,  H1 = 96,  W1 = 80;    // win 7
  const int D2 = 40,  H2 = 48,  W2 = 40;    // win 5

  const size_t n1 = (size_t)N * D1 * H1 * W1;   // 1,228,800
  const size_t n2 = (size_t)N * D2 * H2 * W2;   //   153,600

  float* ws = (float*)d_ws;
  float* I1 = ws; ws += n1;
  float* J1 = ws; ws += n1;
  float* I2 = ws; ws += n2;
  float* J2 = ws; ws += n2;

  const dim3 g0(cdiv(W0, TW), cdiv(H0, TH), N * cdiv(D0, TD));
  const dim3 g1(cdiv(W1, TW), cdiv(H1, TH), N * cdiv(D1, TD));
  const dim3 g2(cdiv(W2, TW), cdiv(H2, TH), N * cdiv(D2, TD));
  const int nb0 = (int)(g0.x * g0.y * g0.z);
  const int nb1 = (int)(g1.x * g1.y * g1.z);
  const int nb2 = (int)(g2.x * g2.y * g2.z);

  float* p0 = ws; ws += nb0;
  float* p1 = ws; ws += nb1;
  float* p2 = ws; ws += nb2;
  // total ws usage ~11.1 MB

  const float cnt0 = (float)((size_t)N * D0 * H0 * W0);
  const float cnt1 = (float)n1;
  const float cnt2 = (float)n2;
  const float c0 = -1.0f / (3.0f * cnt0);
  const float c1 = -1.0f / (3.0f * cnt1);
  const float c2 = -1.0f / (3.0f * cnt2);

  // scale 0 (win 9)
  ncc_scale_kernel<4><<<g0, NCC_NTHREADS, 0, stream>>>(I0, J0, D0, H0, W0,
                                                       (int)cdiv(D0, TD), p0);
  // pyramid level 1
  {
    const int blocks = (int)cdiv((unsigned)n1, NCC_NTHREADS);
    avgpool_kernel<<<blocks, NCC_NTHREADS, 0, stream>>>(I0, I1, D0, H0, W0, D1, H1, W1, N);
    avgpool_kernel<<<blocks, NCC_NTHREADS, 0, stream>>>(J0, J1, D0, H0, W0, D1, H1, W1, N);
  }
  // scale 1 (win 7)
  ncc_scale_kernel<3><<<g1, NCC_NTHREADS, 0, stream>>>(I1, J1, D1, H1, W1,
                                                       (int)cdiv(D1, TD), p1);
  // pyramid level 2
  {
    const int blocks = (int)cdiv((unsigned)n2, NCC_NTHREADS);
    avgpool_kernel<<<blocks, NCC_NTHREADS, 0, stream>>>(I1, I2, D1, H1, W1, D2, H2, W2, N);
    avgpool_kernel<<<blocks, NCC_NTHREADS, 0, stream>>>(J1, J2, D1, H1, W1, D2, H2, W2, N);
  }
  // scale 2 (win 5)
  ncc_scale_kernel<2><<<g2, NCC_NTHREADS, 0, stream>>>(I2, J2, D2, H2, W2,
                                                       (int)cdiv(D2, TD), p2);

  final_reduce_kernel<<<1, NCC_NTHREADS, 0, stream>>>(p0, nb0, p1, nb1, p2, nb2,
                                                      c0, c1, c2, out);
}